// SCEmbed_35373350650017
// MI455X (gfx1250) — compile-verified
//
#include <hip/hip_runtime.h>
#include <stdint.h>

#define BATCH   256
#define SEQ     4096
#define N_GENES 30000
#define DIM     128

#define WAVES          8
#define TILE           16                      // rows per TDM gather (16-bit idx max 16)
#define SEQ_PER_WAVE   (SEQ / WAVES)           // 512
#define TILES_PER_WAVE (SEQ_PER_WAVE / TILE)   // 32
#define TILE_FLTS      (TILE * DIM)            // 2048 floats = 8 KB

typedef unsigned int v4u __attribute__((ext_vector_type(4)));
typedef int          v4i __attribute__((ext_vector_type(4)));
typedef int          v8i __attribute__((ext_vector_type(8)));
typedef float        v2f __attribute__((ext_vector_type(2)));
typedef float        v8f __attribute__((ext_vector_type(8)));

// Issue one TDM gather of 16 embedding rows (u16 indices held in lanes 0..15 of idxu)
// into LDS at byte offset lds_off. Descriptor per CDNA5 ISA.
static __device__ __forceinline__ void tdm_gather16(unsigned lds_off,
                                                    unsigned long long ea,
                                                    v8i g1, unsigned idxu) {
  // pack 16 u16 row indices into 8 scalar dwords (D# groups 2 and 3, gather mode)
  int i0 = __builtin_amdgcn_readlane((int)idxu, 0)  | (__builtin_amdgcn_readlane((int)idxu, 1)  << 16);
  int i1 = __builtin_amdgcn_readlane((int)idxu, 2)  | (__builtin_amdgcn_readlane((int)idxu, 3)  << 16);
  int i2 = __builtin_amdgcn_readlane((int)idxu, 4)  | (__builtin_amdgcn_readlane((int)idxu, 5)  << 16);
  int i3 = __builtin_amdgcn_readlane((int)idxu, 6)  | (__builtin_amdgcn_readlane((int)idxu, 7)  << 16);
  int i4 = __builtin_amdgcn_readlane((int)idxu, 8)  | (__builtin_amdgcn_readlane((int)idxu, 9)  << 16);
  int i5 = __builtin_amdgcn_readlane((int)idxu, 10) | (__builtin_amdgcn_readlane((int)idxu, 11) << 16);
  int i6 = __builtin_amdgcn_readlane((int)idxu, 12) | (__builtin_amdgcn_readlane((int)idxu, 13) << 16);
  int i7 = __builtin_amdgcn_readlane((int)idxu, 14) | (__builtin_amdgcn_readlane((int)idxu, 15) << 16);

  v4u g0;
  g0[0] = 0x80000001u;                                   // count=1, gather_mode=1, 16-bit idx
  g0[1] = lds_off;                                       // lds_addr (bytes)
  g0[2] = (unsigned)ea;                                  // global_addr[31:0]
  g0[3] = ((unsigned)(ea >> 32) & 0x01FFFFFFu) | (2u << 30); // global_addr[56:32], type=2

  v4i g2; g2[0] = i0; g2[1] = i1; g2[2] = i2; g2[3] = i3;    // row_index_0..7
  v4i g3; g3[0] = i4; g3[1] = i5; g3[2] = i6; g3[3] = i7;    // row_index_8..15

  v8i gz = {0, 0, 0, 0, 0, 0, 0, 0};                     // 6-arg form: extra group (unused)

  // make sure prior ds reads of the target buffer retired before TDM overwrites it
  asm volatile("s_wait_dscnt 0" ::: "memory");
  __builtin_amdgcn_tensor_load_to_lds(g0, g1, g2, g3, gz, 0);
}

__global__ __launch_bounds__(WAVES * 32) void scembed_pool_kernel(
    const long long* __restrict__ gids,
    const float*     __restrict__ cnts,
    const float*     __restrict__ emb,
    float*           __restrict__ out)
{
  __shared__ __align__(16) float stage[WAVES][2][TILE_FLTS]; // 128 KB double-buffered gather stage
  __shared__ float P[WAVES][DIM];                            // 4 KB per-wave partials

  const int b    = blockIdx.x;
  const int wave = (int)(threadIdx.x >> 5);
  const int lane = (int)(threadIdx.x & 31);
  const int base = b * SEQ + wave * SEQ_PER_WAVE;

  // D# group1 (static): data_size=4B, tensor=[DIM x N_GENES], tile=[DIM x 16 indices],
  // row stride = DIM elements, no padding / barrier / iterate / multicast.
  v8i g1;
  g1[0] = 0x00020000;                                          // data_size=2 (4 bytes)
  g1[1] = (int)((DIM & 0xFFFF) << 16);                         // tensor_dim0[15:0]
  g1[2] = (int)(((N_GENES & 0xFFFF) << 16) | ((DIM >> 16) & 0xFFFF));   // dim1 lo | dim0 hi
  g1[3] = (int)(((DIM & 0xFFFF) << 16) | ((N_GENES >> 16) & 0xFFFF));   // tile_dim0 | dim1 hi
  g1[4] = TILE;                                                // tile_dim1 = #valid indices
  g1[5] = DIM;                                                 // tensor_dim0_stride lo32
  g1[6] = 0;
  g1[7] = 0;

  const unsigned long long ea = (unsigned long long)(uintptr_t)emb;
  const unsigned stage_off0 = (unsigned)(uintptr_t)&stage[wave][0][0];
  const unsigned stage_off1 = (unsigned)(uintptr_t)&stage[wave][1][0];

  // ---- prologue: load tile 0 indices/weights, issue its gather ----
  unsigned iu; float cm;
  {
    int s = base + (lane & 15);
    long long g = gids[s];
    float c = cnts[s];
    iu = (g < 0 || g >= N_GENES) ? 0u : (unsigned)g;
    cm = (g < 0) ? 0.0f : c;
  }
  tdm_gather16(stage_off0, ea, g1, iu);
  float cur_cnt = cm;

  float4 acc = make_float4(0.f, 0.f, 0.f, 0.f);

  for (int t = 0; t < TILES_PER_WAVE; ++t) {
    if (t + 1 < TILES_PER_WAVE) {
      int s = base + (t + 1) * TILE + (lane & 15);
      long long g = gids[s];
      float c = cnts[s];
      iu = (g < 0 || g >= N_GENES) ? 0u : (unsigned)g;
      cm = (g < 0) ? 0.0f : c;
      tdm_gather16(((t + 1) & 1) ? stage_off1 : stage_off0, ea, g1, iu);
      __builtin_amdgcn_s_wait_tensorcnt(1);   // tile t's gather is complete (in-order)
    } else {
      __builtin_amdgcn_s_wait_tensorcnt(0);
    }
    asm volatile("" ::: "memory");

    const float4* st = (const float4*)&stage[wave][t & 1][0];
#pragma unroll
    for (int k = 0; k < TILE; ++k) {
      float c = __int_as_float(__builtin_amdgcn_readlane(__float_as_int(cur_cnt), k));
      float4 r = st[k * (DIM / 4) + lane];    // ds_load_b128: row k, this lane's 4 dims
      acc.x += c * r.x;
      acc.y += c * r.y;
      acc.z += c * r.z;
      acc.w += c * r.w;
    }
    cur_cnt = cm;
  }

  // per-wave partials -> LDS
  P[wave][4 * lane + 0] = acc.x;
  P[wave][4 * lane + 1] = acc.y;
  P[wave][4 * lane + 2] = acc.z;
  P[wave][4 * lane + 3] = acc.w;
  __syncthreads();

  // ---- cross-wave reduction via WMMA (A = ones, D[0,n] = sum_w P[w][n]) ----
  if (wave == 0) {                          // uniform branch: EXEC all-ones inside
    v2f a; a[0] = 1.0f; a[1] = 1.0f;        // A[m,k] = 1 for all m,k
    const int col   = lane & 15;
    const int rhalf = (lane >> 4) * 2;      // B layout: vgpr0 = K {0|2}, vgpr1 = K {1|3}
#pragma unroll
    for (int c8 = 0; c8 < 8; ++c8) {        // 8 chunks of 16 dims
      v2f bl, bh;
      bl[0] = P[rhalf + 0][c8 * 16 + col];
      bl[1] = P[rhalf + 1][c8 * 16 + col];
      bh[0] = P[4 + rhalf + 0][c8 * 16 + col];
      bh[1] = P[4 + rhalf + 1][c8 * 16 + col];
      v8f d = {};
      d = __builtin_amdgcn_wmma_f32_16x16x4_f32(false, a, false, bl, (short)0, d, false, false);
      d = __builtin_amdgcn_wmma_f32_16x16x4_f32(false, a, false, bh, (short)0, d, false, false);
      if (lane < 16) out[b * DIM + c8 * 16 + lane] = d[0];   // row M=0 of D
    }
  }
}

extern "C" void kernel_launch(void* const* d_in, const int* in_sizes, int n_in,
                              void* d_out, int out_size, void* d_ws, size_t ws_size,
                              hipStream_t stream) {
  (void)in_sizes; (void)n_in; (void)out_size; (void)d_ws; (void)ws_size;
  const long long* gids = (const long long*)d_in[0];
  const float*     cnts = (const float*)d_in[1];
  const float*     emb  = (const float*)d_in[2];
  float*           out  = (float*)d_out;
  scembed_pool_kernel<<<dim3(BATCH), dim3(WAVES * 32), 0, stream>>>(gids, cnts, emb, out);
}